// SGConv_44693429682702
// MI455X (gfx1250) — compile-verified
//
#include <hip/hip_runtime.h>
#include <math.h>

// SGConv on MI455X (gfx1250, wave32).
// Exploits block-tridiagonal structure of the Chebyshev operator:
// out[t] = sum_{dt=-2..2} x[t+dt] @ G[class(t)][dt] + bias,
// with only 7 distinct time-classes of 16x16 stencil matrices.
// Heavy path uses V_WMMA_F32_16X16X4_F32 (full f32 precision).

typedef __attribute__((ext_vector_type(2))) float v2f;
typedef __attribute__((ext_vector_type(8))) float v8f;

#define SEQ 128
#define NFEAT 16
#define NBATCH 1024
#define NCLS 7   // t = 0,1,2, interior(3..124), 125,126,127
#define NTAP 5   // dt = -2..+2

// ---------------------------------------------------------------------------
// Kernel 1: build the 7x5 stencil matrices G (each 16x16 f32) into d_ws.
// Single workgroup; power iteration for ||L||_2 runs entirely in LDS.
// ---------------------------------------------------------------------------
__global__ __launch_bounds__(256) void sg_setup(
    const float* __restrict__ weight,   // [3][16][16]
    const float* __restrict__ adj,      // [16][16]
    float* __restrict__ G)              // [NCLS][NTAP][16][16]
{
  __shared__ float feat[256], L_int[256], L_bnd[256];
  __shared__ float di_int[16], di_bnd[16];
  __shared__ float craw_ii[16], craw_bi[16], chat_ii[16], chat_bi[16];
  __shared__ float D_int[256], D_bnd[256], P_int[256], P_bnd[256];
  __shared__ float Q_ii[256], Q_bi[256], M0[256];
  __shared__ float v[2048], w[2048], red[256], s_nrm;

  const int tid = threadIdx.x;
  const int f = tid >> 4, g = tid & 15;

  // Feature adjacency: sigmoid(adj) with zero diagonal.
  {
    float s = 1.0f / (1.0f + __expf(-adj[tid]));
    feat[tid] = (f == g) ? 0.0f : s;
  }
  __syncthreads();

  // Degrees: interior rows get +3 (self + 2 temporal), boundary +2.
  if (tid < 16) {
    float r = 0.0f;
    for (int k = 0; k < 16; ++k) r += feat[tid * 16 + k];
    di_int[tid] = rsqrtf(r + 3.0f);
    di_bnd[tid] = rsqrtf(r + 2.0f);
  }
  __syncthreads();

  // Laplacian diagonal blocks L = I - D^-1/2 A D^-1/2 (two variants);
  // off-diagonal blocks are -diag(craw_*).
  {
    float Ad = feat[tid] + ((f == g) ? 1.0f : 0.0f);
    L_int[tid] = ((f == g) ? 1.0f : 0.0f) - Ad * di_int[f] * di_int[g];
    L_bnd[tid] = ((f == g) ? 1.0f : 0.0f) - Ad * di_bnd[f] * di_bnd[g];
  }
  if (tid < 16) {
    craw_ii[tid] = di_int[tid] * di_int[tid];
    craw_bi[tid] = di_bnd[tid] * di_int[tid];
  }
  // Deterministic pseudo-random init vector for power iteration.
  for (int i = tid; i < 2048; i += 256) {
    unsigned h = (unsigned)i * 2654435761u;
    v[i] = (float)((h >> 9) & 0xFFFF) * (1.0f / 65536.0f) - 0.5f;
  }
  __syncthreads();

  // Power iteration: sigma = ||L||_2 (L symmetric -> |lambda_max|).
  for (int it = 0; it < 300; ++it) {
    for (int i = tid; i < 2048; i += 256) {
      const int s = i >> 4, ff = i & 15;
      const float* Lb = (s == 0 || s == 127) ? L_bnd : L_int;
      const int base = i & ~15;
      float acc = 0.0f;
      #pragma unroll
      for (int k = 0; k < 16; ++k) acc += Lb[ff * 16 + k] * v[base + k];
      if (s > 0)   acc -= ((s == 1 || s == 127) ? craw_bi[ff] : craw_ii[ff]) * v[i - 16];
      if (s < 127) acc -= ((s == 0 || s == 126) ? craw_bi[ff] : craw_ii[ff]) * v[i + 16];
      w[i] = acc;
    }
    __syncthreads();
    float p = 0.0f;
    for (int i = tid; i < 2048; i += 256) p += w[i] * w[i];
    red[tid] = p;
    __syncthreads();
    for (int off = 128; off > 0; off >>= 1) {
      if (tid < off) red[tid] += red[tid + off];
      __syncthreads();
    }
    if (tid == 0) s_nrm = sqrtf(red[0]);
    __syncthreads();
    const float inv = 1.0f / s_nrm;
    for (int i = tid; i < 2048; i += 256) v[i] = w[i] * inv;
    __syncthreads();
  }

  // T1 = (2/sigma) L - I blocks; chat = off-diagonal 16-vectors of T1.
  const float s2 = 2.0f / s_nrm;
  D_int[tid] = s2 * L_int[tid] - ((f == g) ? 1.0f : 0.0f);
  D_bnd[tid] = s2 * L_bnd[tid] - ((f == g) ? 1.0f : 0.0f);
  if (tid < 16) {
    chat_ii[tid] = -s2 * craw_ii[tid];
    chat_bi[tid] = -s2 * craw_bi[tid];
  }
  __syncthreads();

  const float* W0 = weight;
  const float* W1 = weight + 256;
  const float* W2 = weight + 512;
  // P = W1 + 2 D W2 ; Q = 2 diag(chat) W2 ; M0 = W0 - W2.
  {
    float a = 0.0f, b = 0.0f;
    for (int k = 0; k < 16; ++k) {
      a += D_int[f * 16 + k] * W2[k * 16 + g];
      b += D_bnd[f * 16 + k] * W2[k * 16 + g];
    }
    P_int[tid] = W1[tid] + 2.0f * a;
    P_bnd[tid] = W1[tid] + 2.0f * b;
    Q_ii[tid] = 2.0f * chat_ii[f] * W2[tid];
    Q_bi[tid] = 2.0f * chat_bi[f] * W2[tid];
    M0[tid] = W0[tid] - W2[tid];
  }
  __syncthreads();

  // Build G for 7 representative t values. Thread tid computes element (f,g).
  const int reps[NCLS] = {0, 1, 2, 64, 125, 126, 127};
  for (int ci = 0; ci < NCLS; ++ci) {
    const int t = reps[ci];
    auto Dm = [&](int s) { return (s == 0 || s == 127) ? D_bnd : D_int; };
    auto Pm = [&](int s) { return (s == 0 || s == 127) ? P_bnd : P_int; };
    auto cv = [&](int s) { return (s == 0 || s == 126) ? chat_bi : chat_ii; }; // edge (s,s+1)
    auto Qm = [&](int s) { return (s == 0 || s == 126) ? Q_bi : Q_ii; };       // edge (s,s+1)

    float o5[NTAP] = {0.f, 0.f, 0.f, 0.f, 0.f};
    if (t >= 2) o5[0] = cv(t - 2)[f] * Qm(t - 1)[tid];
    if (t >= 1) {
      float acc = 0.0f;
      const float* A = Dm(t - 1); const float* Bq = Qm(t - 1);
      for (int k = 0; k < 16; ++k) acc += A[f * 16 + k] * Bq[k * 16 + g];
      o5[1] = cv(t - 1)[f] * Pm(t)[tid] + acc;
    }
    {
      float acc = 0.0f;
      const float* A = Dm(t); const float* Bp = Pm(t);
      for (int k = 0; k < 16; ++k) acc += A[f * 16 + k] * Bp[k * 16 + g];
      float r = M0[tid] + acc;
      if (t >= 1)   r += cv(t - 1)[f] * Qm(t - 1)[tid];
      if (t <= 126) r += cv(t)[f] * Qm(t)[tid];
      o5[2] = r;
    }
    if (t <= 126) {
      float acc = 0.0f;
      const float* A = Dm(t + 1); const float* Bq = Qm(t);
      for (int k = 0; k < 16; ++k) acc += A[f * 16 + k] * Bq[k * 16 + g];
      o5[3] = cv(t)[f] * Pm(t)[tid] + acc;
    }
    if (t <= 125) o5[4] = cv(t + 1)[f] * Qm(t)[tid];

    for (int tap = 0; tap < NTAP; ++tap)
      G[(ci * NTAP + tap) * 256 + tid] = o5[tap];
  }
}

// ---------------------------------------------------------------------------
// Kernel 2: stencil of 16x16 f32 WMMAs. One wave32 per (16-batch, 8-t) tile.
// ---------------------------------------------------------------------------
__device__ __forceinline__ void load_xtile(const float* __restrict__ xrow,
                                           int t, int sel, v2f* dst) {
  if ((unsigned)t < (unsigned)SEQ) {
    const float* p = xrow + t * NFEAT + sel;
    #pragma unroll
    for (int c = 0; c < 4; ++c) dst[c] = *(const v2f*)(p + 4 * c);
  } else {
    #pragma unroll
    for (int c = 0; c < 4; ++c) { v2f z; z.x = 0.f; z.y = 0.f; dst[c] = z; }
  }
}

__global__ __launch_bounds__(32) void sg_main(
    const float* __restrict__ x,     // [1024][128][16]
    const float* __restrict__ G,     // [NCLS][NTAP][16][16]
    const float* __restrict__ bias,  // [16]
    float* __restrict__ out)         // [1024][128][16]
{
  const int bt = blockIdx.x >> 4;   // batch tile 0..63
  const int tc = blockIdx.x & 15;   // time chunk 0..15 (8 steps each)
  const int t0 = tc * 8;
  const int lane = threadIdx.x;
  const int row = lane & 15;        // A-frag M / B-frag N / D-frag N
  const int hi = lane >> 4;
  const int sel = hi * 2;           // K sub-select per lane half
  const float* xrow = x + (size_t)(bt * 16 + row) * (SEQ * NFEAT);

  // Interior-class (cls 3) stencil matrices as B fragments in registers.
  v2f gi[NTAP][4];
  #pragma unroll
  for (int tap = 0; tap < NTAP; ++tap) {
    const float* M = G + (3 * NTAP + tap) * 256;
    #pragma unroll
    for (int c = 0; c < 4; ++c) {
      const int k = 4 * c + sel;
      v2f t2; t2.x = M[k * 16 + row]; t2.y = M[(k + 1) * 16 + row];
      gi[tap][c] = t2;
    }
  }
  const float bfrag = bias[row];    // D-layout: element (M,N) = bias[N]

  // Rolling x fragments: xf[0..4] hold tiles t-2..t+2.
  v2f xf[NTAP][4];
  load_xtile(xrow, t0 - 2, sel, xf[0]);
  load_xtile(xrow, t0 - 1, sel, xf[1]);
  load_xtile(xrow, t0 + 0, sel, xf[2]);
  load_xtile(xrow, t0 + 1, sel, xf[3]);

  for (int ti = 0; ti < 8; ++ti) {
    const int t = t0 + ti;
    load_xtile(xrow, t + 2, sel, xf[4]);
    if ((unsigned)(t + 3) < (unsigned)SEQ)
      __builtin_prefetch(xrow + (t + 3) * NFEAT, 0, 1);

    v8f acc = {bfrag, bfrag, bfrag, bfrag, bfrag, bfrag, bfrag, bfrag};
    const int cls = (t >= 3 && t <= 124) ? 3 : ((t < 3) ? t : t - 121);

    if (cls == 3) {
      // 20 chained V_WMMA_F32_16X16X4_F32 per output tile.
      #pragma unroll
      for (int tap = 0; tap < NTAP; ++tap) {
        #pragma unroll
        for (int c = 0; c < 4; ++c)
          acc = __builtin_amdgcn_wmma_f32_16x16x4_f32(
              false, xf[tap][c], false, gi[tap][c], (short)0, acc, false, false);
      }
    } else {
      // Edge time steps (6 of 128): fetch the class's matrices on the fly.
      for (int tap = 0; tap < NTAP; ++tap) {
        const int tt = t + tap - 2;
        if ((unsigned)tt >= (unsigned)SEQ) continue;   // wave-uniform
        const float* M = G + (cls * NTAP + tap) * 256;
        #pragma unroll
        for (int c = 0; c < 4; ++c) {
          const int k = 4 * c + sel;
          v2f gb; gb.x = M[k * 16 + row]; gb.y = M[(k + 1) * 16 + row];
          acc = __builtin_amdgcn_wmma_f32_16x16x4_f32(
              false, xf[tap][c], false, gb, (short)0, acc, false, false);
        }
      }
    }

    // Store D-fragment: lane holds (M = r + 8*hi, N = row) for r = 0..7.
    float* op = out + (size_t)(bt * 16) * (SEQ * NFEAT) + (size_t)t * NFEAT;
    #pragma unroll
    for (int r = 0; r < 8; ++r)
      op[(size_t)(r + 8 * hi) * (SEQ * NFEAT) + row] = acc[r];

    // Shift rolling window.
    #pragma unroll
    for (int c = 0; c < 4; ++c) {
      xf[0][c] = xf[1][c]; xf[1][c] = xf[2][c];
      xf[2][c] = xf[3][c]; xf[3][c] = xf[4][c];
    }
  }
}

// ---------------------------------------------------------------------------
extern "C" void kernel_launch(void* const* d_in, const int* in_sizes, int n_in,
                              void* d_out, int out_size, void* d_ws, size_t ws_size,
                              hipStream_t stream) {
  (void)in_sizes; (void)n_in; (void)out_size; (void)ws_size;
  const float* x      = (const float*)d_in[0];  // [1024,128,16]
  const float* weight = (const float*)d_in[1];  // [3,16,16]
  const float* bias   = (const float*)d_in[2];  // [16]
  const float* adj    = (const float*)d_in[3];  // [16,16]
  float* G = (float*)d_ws;                      // NCLS*NTAP*256 floats = 35 KB

  sg_setup<<<1, 256, 0, stream>>>(weight, adj, G);
  sg_main<<<64 * 16, 32, 0, stream>>>(x, G, bias, (float*)d_out);
}